// _GATEdgeHead_87342454931783
// MI455X (gfx1250) — compile-verified
//
#include <hip/hip_runtime.h>
#include <hip/hip_bf16.h>

typedef float v2f __attribute__((ext_vector_type(2)));
typedef float v4f __attribute__((ext_vector_type(4)));
typedef float v8f __attribute__((ext_vector_type(8)));

#define EDIM 256
#define NSEQ 2048
#define BATCH 4
#define NROWS (BATCH * NSEQ)   // 8192
#define EPSV 1e-5f
#define SLOPE 0.2f

// ---------------------------------------------------------------------------
// Kernel 1: fold the big GEMM through the rank-4 projections.
// P[n][f] = sum_e a_sel(n,e) * W[e][f],  n=0..3 -> a_left rows, 4..7 -> a_right
// ---------------------------------------------------------------------------
__global__ void __launch_bounds__(256)
proj_fold_kernel(const float* __restrict__ W, const float* __restrict__ a,
                 float* __restrict__ P) {
    const int n = blockIdx.x;        // 0..7
    const int f = threadIdx.x;       // 0..255 (coalesced over W columns)
    const int arow  = (n < 4) ? n : (n - 4);
    const int acol0 = (n < 4) ? 0 : EDIM;
    float acc = 0.0f;
#pragma unroll 8
    for (int e = 0; e < EDIM; ++e) {
        acc = fmaf(a[arow * (2 * EDIM) + acol0 + e], W[e * EDIM + f], acc);
    }
    P[n * EDIM + f] = acc;
}

// ---------------------------------------------------------------------------
// Kernel 2: fused LayerNorm + skinny projection via v_wmma_f32_16x16x4_f32.
// Block = 64 threads (2 waves); each wave owns one 16-row tile.
// S[row][0..3] = s_i(k),  S[row][4..7] = s_j(k)
// ---------------------------------------------------------------------------
__global__ void __launch_bounds__(64)
ln_proj_wmma_kernel(const float* __restrict__ h, const float* __restrict__ gamma,
                    const float* __restrict__ beta, const float* __restrict__ P,
                    float* __restrict__ S) {
    __shared__ float tiles[2][16 * EDIM];   // 2 x 16KB normalized h tiles
    __shared__ float Pt16[EDIM * 16];       // 16KB: Pt16[k*16+n] = (n<8)?P[n][k]:0

    const int tid  = threadIdx.x;
    const int wave = tid >> 5;
    const int lane = tid & 31;
    const int m    = lane & 15;        // row within tile / N column
    const int half = lane >> 4;        // 0: cols 0-127, 1: cols 128-255
    const int row0 = blockIdx.x * 32 + wave * 16;
    const int row  = row0 + m;

    // Zero-padded transposed P into LDS (masks N=8..15 so EXEC stays uniform
    // at the WMMA).
    for (int idx = tid; idx < EDIM * 16; idx += 64) {
        const int k = idx >> 4, n = idx & 15;
        Pt16[idx] = (n < 8) ? P[n * EDIM + k] : 0.0f;
    }

    // --- Phase 1: LayerNorm. Lane pair (m, m+16) covers one row. ---
    float* T = &tiles[wave][0];
    const float* hp = h + (size_t)row * EDIM + half * 128;
    float sum = 0.0f, sumsq = 0.0f;
#pragma unroll
    for (int c4 = 0; c4 < 32; ++c4) {
        const v4f v = ((const v4f*)hp)[c4];
        ((v4f*)&T[m * EDIM + half * 128])[c4] = v;
        sum   += v[0] + v[1] + v[2] + v[3];
        sumsq += v[0]*v[0] + v[1]*v[1] + v[2]*v[2] + v[3]*v[3];
    }
    sum   += __shfl_xor(sum, 16);
    sumsq += __shfl_xor(sumsq, 16);
    const float mu   = sum * (1.0f / EDIM);
    const float var  = sumsq * (1.0f / EDIM) - mu * mu;
    const float rstd = rsqrtf(var + EPSV);

    const v4f* g4 = (const v4f*)(gamma + half * 128);
    const v4f* b4 = (const v4f*)(beta  + half * 128);
#pragma unroll
    for (int c4 = 0; c4 < 32; ++c4) {
        v4f v = ((v4f*)&T[m * EDIM + half * 128])[c4];
        const v4f gg = g4[c4], bb = b4[c4];
#pragma unroll
        for (int k = 0; k < 4; ++k)
            v[k] = (v[k] - mu) * rstd * gg[k] + bb[k];
        ((v4f*)&T[m * EDIM + half * 128])[c4] = v;
    }
    __syncthreads();

    // --- Phase 2: S_tile(16x16) = T(16x256) @ Pt(256x16) via 64 f32 WMMAs ---
    // A layout (16x4 f32): lanes 0-15 hold K={kk,kk+1}, lanes 16-31 K={kk+2,kk+3}
    v8f acc = {0.f,0.f,0.f,0.f,0.f,0.f,0.f,0.f};
#pragma unroll 4
    for (int kk = 0; kk < EDIM; kk += 4) {
        const int kb = kk + half * 2;
        const v2f A = *(const v2f*)&T[m * EDIM + kb];      // 8B-aligned ds_load_b64
        v2f Bv;
        Bv[0] = Pt16[kb * 16 + m];
        Bv[1] = Pt16[(kb + 1) * 16 + m];
        acc = __builtin_amdgcn_wmma_f32_16x16x4_f32(
            /*neg_a=*/false, A, /*neg_b=*/false, Bv,
            /*c_mod=*/(short)0, acc, /*reuse_a=*/false, /*reuse_b=*/false);
    }

    // D layout: VGPR r -> (M = r + half*8, N = m); only N<8 carries real data.
    if (m < 8) {
#pragma unroll
        for (int r = 0; r < 8; ++r)
            S[(size_t)(row0 + half * 8 + r) * 8 + m] = acc[r];
    }
}

// ---------------------------------------------------------------------------
// Kernel 3: pairwise symmetric leaky-ReLU epilogue (bandwidth bound, 67 MB out)
// out[b,i,j] = 0.125 * sum_k [ lrelu(si[i,k]+sj[j,k]) + lrelu(si[j,k]+sj[i,k]) ]
// 64x64 tile per block; each thread emits a 4x4 patch with float4 stores.
// ---------------------------------------------------------------------------
__device__ __forceinline__ float lrelu(float x) { return fmaxf(x, SLOPE * x); }

__global__ void __launch_bounds__(256)
pairwise_kernel(const float* __restrict__ S, float* __restrict__ out) {
    __shared__ float sI[64 * 8];
    __shared__ float sJ[64 * 8];
    const int b  = blockIdx.z;
    const int i0 = blockIdx.y * 64;
    const int j0 = blockIdx.x * 64;
    const int t  = threadIdx.x;

    const float* Sb = S + (size_t)b * NSEQ * 8;
    sI[t]       = Sb[i0 * 8 + t];
    sI[t + 256] = Sb[i0 * 8 + t + 256];
    sJ[t]       = Sb[j0 * 8 + t];
    sJ[t + 256] = Sb[j0 * 8 + t + 256];
    __syncthreads();

    const int tx = t & 15;    // j sub-tile
    const int ty = t >> 4;    // i sub-tile

    v4f siJ[4], sjJ[4];
#pragma unroll
    for (int c = 0; c < 4; ++c) {
        siJ[c] = *(const v4f*)&sJ[(tx * 4 + c) * 8];       // s_i of row j
        sjJ[c] = *(const v4f*)&sJ[(tx * 4 + c) * 8 + 4];   // s_j of row j
    }

#pragma unroll
    for (int aa = 0; aa < 4; ++aa) {
        const v4f siA = *(const v4f*)&sI[(ty * 4 + aa) * 8];      // s_i of row i
        const v4f sjA = *(const v4f*)&sI[(ty * 4 + aa) * 8 + 4];  // s_j of row i
        v4f r;
#pragma unroll
        for (int c = 0; c < 4; ++c) {
            float acc = 0.0f;
#pragma unroll
            for (int k = 0; k < 4; ++k) {
                acc += lrelu(siA[k] + sjJ[c][k]);   // logits[b,i,j] term
                acc += lrelu(siJ[c][k] + sjA[k]);   // logits[b,j,i] term
            }
            r[c] = 0.125f * acc;                    // 0.5 * mean_k(...)
        }
        float* orow = out + ((size_t)b * NSEQ + (i0 + ty * 4 + aa)) * NSEQ
                          + j0 + tx * 4;
        *(v4f*)orow = r;                            // coalesced global_store_b128
    }
}

// ---------------------------------------------------------------------------
extern "C" void kernel_launch(void* const* d_in, const int* in_sizes, int n_in,
                              void* d_out, int out_size, void* d_ws, size_t ws_size,
                              hipStream_t stream) {
    (void)in_sizes; (void)n_in; (void)out_size; (void)ws_size;
    const float* h     = (const float*)d_in[0];
    const float* gamma = (const float*)d_in[1];
    const float* beta  = (const float*)d_in[2];
    const float* W     = (const float*)d_in[3];
    const float* a     = (const float*)d_in[4];
    float* out = (float*)d_out;

    float* P = (float*)d_ws;         // 8 x 256 folded projection
    float* S = P + 8 * EDIM;         // 8192 x 8 fused scores (si||sj)

    proj_fold_kernel<<<dim3(8), dim3(256), 0, stream>>>(W, a, P);
    ln_proj_wmma_kernel<<<dim3(NROWS / 32), dim3(64), 0, stream>>>(h, gamma, beta, P, S);
    pairwise_kernel<<<dim3(NSEQ / 64, NSEQ / 64, BATCH), dim3(256), 0, stream>>>(S, out);
}